// RegularGrid_36112085025320
// MI455X (gfx1250) — compile-verified
//
#include <hip/hip_runtime.h>
#include <stdint.h>

// ---------------- problem constants ----------------
namespace {
constexpr int   kRes     = 128;
constexpr int   kCh      = 28;
constexpr int   kRays    = 4096;
constexpr int   kIntr    = 768;                 // RES*3*2
constexpr int   kT       = 767;                 // samples per ray (t = intersections[:, :-1])
constexpr int   kPts     = kRays * kT;          // 3,141,632
constexpr int   kBlk     = 256;
constexpr int   kNBlk    = kPts / kBlk;         // 12272 (exact)
constexpr int   kIntrPts = kRays * kIntr;       // 3,145,728 (= 12288 * 256, exact)
constexpr int   kCS      = kRes * kRes * kRes;  // channel stride, elems (2,097,152)
constexpr int   kZS      = kRes * kRes;         // 16384
constexpr int   kYS      = kRes;                // 128
constexpr float kVoxelLen = 0.0045105489784345755f; // 2*sqrt(3)/768 (fp32)
constexpr float kNear = 0.5f;
constexpr float kFar  = 6.0f;
} // namespace

// ---------------- CDNA5 async global->LDS gather ----------------
// Per-lane gather: copies 4 bytes from global address sbase plus voff to the LDS address in lds_off; tracked by ASYNCcnt.
#define ASYNC_GATHER_B32(lds_off, voff, sbase)                         \
  asm volatile("global_load_async_to_lds_b32 %0, %1, %2"               \
               :: "v"(lds_off), "v"(voff), "s"(sbase) : "memory")

#define WAIT_ASYNCCNT(n)                                               \
  asm volatile("s_wait_asynccnt %0" :: "n"(n) : "memory")

static __device__ __forceinline__ uint32_t lds_off_of(const void* p) {
  // Generic (flat) pointers to __shared__ carry the LDS byte offset in the
  // low 32 bits (aperture form); that is the DS-space address the async
  // instruction's LDS-address VGPR expects.
  return (uint32_t)(uintptr_t)p;
}

// ---------------- helpers ----------------
static __device__ __forceinline__ float ray_entry_t(const float* __restrict__ ro,
                                                    const float* __restrict__ rd,
                                                    int r) {
  float mx = -3.0e38f;
#pragma unroll
  for (int a = 0; a < 3; ++a) {
    float o = ro[r * 3 + a];
    float d = rd[r * 3 + a];
    float op = ( 1.0f - o) / d;
    float on = (-1.0f - o) / d;
    mx = fmaxf(mx, fminf(op, on));
  }
  return fminf(fmaxf(mx, kNear), kFar);   // clip(max(off_in), NEAR, FAR)
}

// ---------------- K0: per-ray AABB entry t ----------------
__global__ void k_ray_start(const float* __restrict__ ro,
                            const float* __restrict__ rd,
                            float* __restrict__ tstart) {
  int r = blockIdx.x * blockDim.x + threadIdx.x;
  if (r < kRays) tstart[r] = ray_entry_t(ro, rd, r);
}

// ---------------- K1: intersections [4096, 768] ----------------
__global__ __launch_bounds__(kBlk) void k_intersections(const float* __restrict__ tstart,
                                                        float* __restrict__ inter) {
  int idx = blockIdx.x * kBlk + threadIdx.x;       // grid is exact
  int r = idx / kIntr;
  int i = idx - r * kIntr;
  inter[idx] = tstart[r] + (float)i * kVoxelLen;
}

// ---------------- K2: mask (as f32) + per-block masked counts ----------------
__global__ __launch_bounds__(kBlk) void k_mask_count(const float* __restrict__ ro,
                                                     const float* __restrict__ rd,
                                                     const float* __restrict__ tstart,
                                                     float* __restrict__ maskf,
                                                     unsigned* __restrict__ counts) {
  __shared__ unsigned wc[kBlk / 32];
  int idx = blockIdx.x * kBlk + threadIdx.x;       // grid is exact
  int r = idx / kT;
  int i = idx - r * kT;
  float t  = tstart[r] + (float)i * kVoxelLen;
  float px = fmaf(t, rd[r * 3 + 0], ro[r * 3 + 0]);
  float py = fmaf(t, rd[r * 3 + 1], ro[r * 3 + 1]);
  float pz = fmaf(t, rd[r * 3 + 2], ro[r * 3 + 2]);
  bool m = (px > -1.0f) && (px < 1.0f) &&
           (py > -1.0f) && (py < 1.0f) &&
           (pz > -1.0f) && (pz < 1.0f);
  maskf[idx] = m ? 1.0f : 0.0f;

  unsigned b    = __builtin_amdgcn_ballot_w32(m);  // wave32 ballot
  unsigned lane = threadIdx.x & 31u;
  unsigned wid  = threadIdx.x >> 5;
  if (lane == 0) wc[wid] = (unsigned)__popc(b);
  __syncthreads();
  if (threadIdx.x == 0) {
    unsigned s = 0;
#pragma unroll
    for (int w = 0; w < kBlk / 32; ++w) s += wc[w];
    counts[blockIdx.x] = s;
  }
}

// ---------------- K3: exclusive scan of block counts (single workgroup) ----------------
__global__ __launch_bounds__(kBlk) void k_scan(const unsigned* __restrict__ counts,
                                               unsigned* __restrict__ offs) {
  __shared__ unsigned tmp[kBlk];
  __shared__ unsigned s_carry;
  if (threadIdx.x == 0) s_carry = 0u;
  __syncthreads();
  for (int base = 0; base < kNBlk; base += kBlk) {
    int g = base + (int)threadIdx.x;
    unsigned v  = (g < kNBlk) ? counts[g] : 0u;
    unsigned c0 = s_carry;
    tmp[threadIdx.x] = v;
    __syncthreads();
#pragma unroll
    for (int off = 1; off < kBlk; off <<= 1) {
      unsigned x = tmp[threadIdx.x];
      if ((int)threadIdx.x >= off) x += tmp[threadIdx.x - off];
      __syncthreads();
      tmp[threadIdx.x] = x;
      __syncthreads();
    }
    unsigned incl = tmp[threadIdx.x];
    if (g < kNBlk) offs[g] = c0 + incl - v;        // exclusive
    __syncthreads();
    if (threadIdx.x == 0) s_carry = c0 + tmp[kBlk - 1];
    __syncthreads();
  }
}

// ---------------- K4: compacted trilinear gather via async LDS pipeline ----------------
__global__ __launch_bounds__(kBlk) void k_gather(const float* __restrict__ ro,
                                                 const float* __restrict__ rd,
                                                 const float* __restrict__ tstart,
                                                 const float* __restrict__ grid,
                                                 const unsigned* __restrict__ blockOff,
                                                 float* __restrict__ out) {
  __shared__ __align__(16) float stage[2][kBlk][8];   // 16 KB double buffer
  __shared__ unsigned wc[kBlk / 32];

  const int tid = threadIdx.x;
  const int idx = blockIdx.x * kBlk + tid;            // grid is exact
  const int r = idx / kT;
  const int i = idx - r * kT;

  const float t  = tstart[r] + (float)i * kVoxelLen;
  const float px = fmaf(t, rd[r * 3 + 0], ro[r * 3 + 0]);
  const float py = fmaf(t, rd[r * 3 + 1], ro[r * 3 + 1]);
  const float pz = fmaf(t, rd[r * 3 + 2], ro[r * 3 + 2]);
  const bool m = (px > -1.0f) && (px < 1.0f) &&
                 (py > -1.0f) && (py < 1.0f) &&
                 (pz > -1.0f) && (pz < 1.0f);

  // ---- deterministic compaction rank (flat order); uniform barrier use ----
  const unsigned b = __builtin_amdgcn_ballot_w32(m);
  unsigned lane = (unsigned)tid & 31u;
  unsigned wid  = (unsigned)tid >> 5;
  if (lane == 0) wc[wid] = (unsigned)__popc(b);
  __syncthreads();
  unsigned prev = 0;
  for (unsigned w = 0; w < wid; ++w) prev += wc[w];
  const unsigned row =
      blockOff[blockIdx.x] + prev + (unsigned)__popc(b & ((1u << lane) - 1u));

  // Wave-uniform early-out: whole wave outside the AABB -> no gathers, no
  // stores. Big traffic saver since consecutive samples map to consecutive
  // lanes. Surviving waves run the async pipeline with EXEC all-ones.
  if (b == 0u) return;

  // ---- voxel coords & trilinear weights (aabb=[-1,1] => pts_n == pts) ----
  const float xs = (px + 1.0f) * 63.5f;   // 0.5*(W-1)
  const float ys = (py + 1.0f) * 63.5f;
  const float zs = (pz + 1.0f) * 63.5f;
  const float x0f = fminf(fmaxf(floorf(xs), 0.0f), 126.0f);
  const float y0f = fminf(fmaxf(floorf(ys), 0.0f), 126.0f);
  const float z0f = fminf(fmaxf(floorf(zs), 0.0f), 126.0f);
  const int x0 = (int)x0f, y0 = (int)y0f, z0 = (int)z0f;
  const float wx = xs - x0f, wy = ys - y0f, wz = zs - z0f;
  const float ux = 1.0f - wx, uy = 1.0f - wy, uz = 1.0f - wz;
  const float w000 = uz * uy * ux, w001 = uz * uy * wx;
  const float w010 = uz * wy * ux, w011 = uz * wy * wx;
  const float w100 = wz * uy * ux, w101 = wz * uy * wx;
  const float w110 = wz * wy * ux, w111 = wz * wy * wx;

  const unsigned vbase = (unsigned)z0 * (unsigned)kZS +
                         (unsigned)y0 * (unsigned)kYS + (unsigned)x0;
  const unsigned dk[8] = {0u, 1u,
                          (unsigned)kYS,         (unsigned)(kYS + 1),
                          (unsigned)kZS,         (unsigned)(kZS + 1),
                          (unsigned)(kZS + kYS), (unsigned)(kZS + kYS + 1)};

  const uint32_t lds0 = lds_off_of(&stage[0][tid][0]);
  const uint32_t lds1 = lds_off_of(&stage[1][tid][0]);

  auto issue = [&](int c, int buf) {
    const unsigned cb = (unsigned)c * (unsigned)kCS + vbase; // elem offset
    const uint32_t lb = buf ? lds1 : lds0;
#pragma unroll
    for (int k = 0; k < 8; ++k) {
      ASYNC_GATHER_B32(lb + 4u * (uint32_t)k, (cb + dk[k]) * 4u, grid);
    }
  };
  auto fold = [&](int buf) -> float {
    const float4* s4 = reinterpret_cast<const float4*>(&stage[buf][tid][0]);
    const float4 lo = s4[0];             // ds_load_b128
    const float4 hi = s4[1];             // ds_load_b128
    float a = lo.x * w000;
    a = fmaf(lo.y, w001, a);
    a = fmaf(lo.z, w010, a);
    a = fmaf(lo.w, w011, a);
    a = fmaf(hi.x, w100, a);
    a = fmaf(hi.y, w101, a);
    a = fmaf(hi.z, w110, a);
    a = fmaf(hi.w, w111, a);
    return a;
  };

  const size_t orow = (size_t)row * (size_t)kCh;

  // Software pipeline: stage channel c+1 through the async unit while
  // folding channel c out of LDS. ASYNCcnt drains in issue order, so
  // waiting <=8 guarantees the older stage's 8 gathers have landed.
  issue(0, 0);
  for (int c = 0; c < kCh - 1; ++c) {
    issue(c + 1, (c + 1) & 1);
    WAIT_ASYNCCNT(8);
    float v = fold(c & 1);
    if (m) out[orow + (size_t)c] = v;
  }
  WAIT_ASYNCCNT(0);
  float v = fold((kCh - 1) & 1);
  if (m) out[orow + (size_t)(kCh - 1)] = v;
}

// ---------------- launcher ----------------
extern "C" void kernel_launch(void* const* d_in, const int* in_sizes, int n_in,
                              void* d_out, int out_size, void* d_ws, size_t ws_size,
                              hipStream_t stream) {
  (void)in_sizes; (void)n_in; (void)ws_size;
  const float* ro   = (const float*)d_in[0];   // rays_o [4096,3]
  const float* rd   = (const float*)d_in[1];   // rays_d [4096,3]
  const float* grid = (const float*)d_in[2];   // data   [28,128,128,128]
  float* out = (float*)d_out;

  // Output layout: [data_interp n_sel*28][mask 4096*767][intersections 4096*768]
  const int n_sel = (out_size - kRays * kT - kRays * kIntr) / kCh;
  float* mask_out  = out + (size_t)n_sel * (size_t)kCh;
  float* inter_out = mask_out + (size_t)kRays * (size_t)kT;

  float*    tstart = (float*)d_ws;
  unsigned* counts = (unsigned*)((char*)d_ws + (size_t)kRays * sizeof(float));
  unsigned* offs   = counts + kNBlk;

  k_ray_start    <<<kRays / kBlk,    kBlk, 0, stream>>>(ro, rd, tstart);
  k_intersections<<<kIntrPts / kBlk, kBlk, 0, stream>>>(tstart, inter_out);
  k_mask_count   <<<kNBlk,           kBlk, 0, stream>>>(ro, rd, tstart, mask_out, counts);
  k_scan         <<<1,               kBlk, 0, stream>>>(counts, offs);
  k_gather       <<<kNBlk,           kBlk, 0, stream>>>(ro, rd, tstart, grid, offs, out);
}